// Encoder_62139586838762
// MI455X (gfx1250) — compile-verified
//
#include <hip/hip_runtime.h>

typedef float v2f __attribute__((ext_vector_type(2)));
typedef float v8f __attribute__((ext_vector_type(8)));

#define K3 125

__device__ __forceinline__ v8f wmma_f32x4(v2f a, v2f b, v8f c) {
    // V_WMMA_F32_16X16X4_F32: D = A(16x4) * B(4x16) + C(16x16)
    return __builtin_amdgcn_wmma_f32_16x16x4_f32(
        false, a, false, b, (short)0, c, false, false);
}

// ---------------------------------------------------------------------------
// degree[dst] += 1 per edge
__global__ void degree_kernel(const int* __restrict__ ei, float* __restrict__ deg, int E) {
    int e = blockIdx.x * blockDim.x + threadIdx.x;
    if (e < E) atomicAdd(&deg[ei[E + e]], 1.0f);
}

// ---------------------------------------------------------------------------
// Scatter: A[dst, kidx, c] += wprod * x[src, c] for the 8 active spline corners.
// One thread per (edge, channel).
__global__ void scatter_edges(const float* __restrict__ x, const int* __restrict__ ei,
                              const float* __restrict__ ea, float* __restrict__ A,
                              int E, int Cin) {
    int tid = blockIdx.x * blockDim.x + threadIdx.x;
    if (tid >= E * Cin) return;
    int e = tid / Cin;
    int c = tid - e * Cin;
    int src = ei[e];
    int dst = ei[E + e];
    float xs = x[(size_t)src * Cin + c];
    float f0, f1, f2;
    int k0, k1, k2;
    {
        float p0 = ea[(size_t)e * 3 + 0] * 4.0f;
        float p1 = ea[(size_t)e * 3 + 1] * 4.0f;
        float p2 = ea[(size_t)e * 3 + 2] * 4.0f;
        float c0 = fminf(fmaxf(floorf(p0), 0.0f), 3.0f);
        float c1 = fminf(fmaxf(floorf(p1), 0.0f), 3.0f);
        float c2 = fminf(fmaxf(floorf(p2), 0.0f), 3.0f);
        k0 = (int)c0; k1 = (int)c1; k2 = (int)c2;
        f0 = p0 - c0; f1 = p1 - c1; f2 = p2 - c2;
    }
#pragma unroll
    for (int b = 0; b < 8; ++b) {
        float w = ((b & 1) ? f0 : 1.0f - f0)
                * ((b & 2) ? f1 : 1.0f - f1)
                * ((b & 4) ? f2 : 1.0f - f2);
        int kidx = (k0 + (b & 1)) + 5 * (k1 + ((b >> 1) & 1)) + 25 * (k2 + ((b >> 2) & 1));
        atomicAdd(&A[((size_t)dst * K3 + kidx) * Cin + c], w * xs);
    }
}

// ---------------------------------------------------------------------------
// h = elu( (A_flat @ W_flat) / max(deg,1) + x @ root + bias )
// A_flat: [N, 125*Cin] row-major, W_flat: [125*Cin, Cout] row-major.
// One wave32 per 16x16 output tile. Branchless loads: ragged M rows are
// address-clamped (their results are dropped at the store), K tails zero B.
__global__ void __launch_bounds__(32)
spline_gemm(const float* __restrict__ A, const float* __restrict__ W,
            const float* __restrict__ x, const float* __restrict__ R,
            const float* __restrict__ bias, const float* __restrict__ deg,
            float* __restrict__ h, int N, int Cin, int Cout) {
    int lane = threadIdx.x;
    int half = lane >> 4;        // selects K pair {0,1} vs {2,3}
    int lm   = lane & 15;
    int m_base = blockIdx.x * 16;
    int n_base = blockIdx.y * 16;
    int m  = m_base + lm;
    int n  = n_base + lm;
    int mc = min(m, N - 1);       // clamped row: OOB rows compute garbage, never stored
    int Kd = K3 * Cin;
    int Kmain = Kd & ~3;

    const float* Arow = A + (size_t)mc * Kd;
    v8f acc = {};
#pragma unroll 2
    for (int k = 0; k < Kmain; k += 4) {
        int ka = k + half * 2;
        v2f a = *(const v2f*)(Arow + ka);                 // global_load_b64
        v2f b;
        b.x = W[(size_t)ka * Cout + n];
        b.y = W[(size_t)(ka + 1) * Cout + n];
        acc = wmma_f32x4(a, b, acc);
    }
    if (Kmain < Kd) {                                     // Kd % 4 == 2 (layer 0 only)
        int ka  = Kmain + half * 2;
        int ka0 = min(ka,     Kd - 1);
        int ka1 = min(ka + 1, Kd - 1);
        v2f a, b;
        a.x = Arow[ka0];
        a.y = Arow[ka1];
        float b0 = W[(size_t)ka0 * Cout + n];
        float b1 = W[(size_t)ka1 * Cout + n];
        b.x = (ka     < Kd) ? b0 : 0.0f;                  // zero B kills OOB products
        b.y = (ka + 1 < Kd) ? b1 : 0.0f;
        acc = wmma_f32x4(a, b, acc);
    }
    // mean aggregation: divide message part by degree (per C-tile row)
#pragma unroll
    for (int i = 0; i < 8; ++i) {
        int mm = min(m_base + i + half * 8, N - 1);
        acc[i] = acc[i] / fmaxf(deg[mm], 1.0f);
    }
    // accumulate root-weight term: x[N,Cin] @ R[Cin,Cout]
    {
        const float* xrow = x + (size_t)mc * Cin;
        int Cmain = Cin & ~3;
        for (int k = 0; k < Cmain; k += 4) {
            int ka = k + half * 2;
            v2f a = *(const v2f*)(xrow + ka);
            v2f b;
            b.x = R[(size_t)ka * Cout + n];
            b.y = R[(size_t)(ka + 1) * Cout + n];
            acc = wmma_f32x4(a, b, acc);
        }
        if (Cmain < Cin) {                                // Cin == 2 (layer 0 only)
            int ka  = Cmain + half * 2;
            int ka0 = min(ka,     Cin - 1);
            int ka1 = min(ka + 1, Cin - 1);
            v2f a, b;
            a.x = xrow[ka0];
            a.y = xrow[ka1];
            float b0 = R[(size_t)ka0 * Cout + n];
            float b1 = R[(size_t)ka1 * Cout + n];
            b.x = (ka     < Cin) ? b0 : 0.0f;
            b.y = (ka + 1 < Cin) ? b1 : 0.0f;
            acc = wmma_f32x4(a, b, acc);
        }
    }
    float bn = bias[n];
#pragma unroll
    for (int i = 0; i < 8; ++i) {
        int mm = m_base + i + half * 8;
        if (mm < N) {
            float v = acc[i] + bn;
            v = (v > 0.0f) ? v : (__expf(v) - 1.0f);      // ELU
            h[(size_t)mm * Cout + n] = v;
        }
    }
}

// ---------------------------------------------------------------------------
// xn[j, c] = sum_i P[i, j] * h[i, c]   (P^T @ h), split-K with f32 atomics.
// P: [Nl, Nl1] row-major, h: [Nl, C]. One wave computes a 64x16 tile (4 accs)
// so B is reused 4x and the 16-lane P reads form 4 adjacent 64B segments.
// Nl and kchunk are multiples of 4 -> the K loop is guard-free.
__global__ void __launch_bounds__(32)
pool_gemm(const float* __restrict__ P, const float* __restrict__ h,
          float* __restrict__ xn, int Nl, int Nl1, int C, int kchunk) {
    int lane = threadIdx.x;
    int half = lane >> 4;
    int lm   = lane & 15;
    int m_base = blockIdx.x * 64;    // rows of xn (cols of P)
    int n_base = blockIdx.y * 16;    // feature cols
    int n = n_base + lm;
    int k0 = blockIdx.z * kchunk;
    int k1 = min(Nl, k0 + kchunk);   // (k1 - k0) % 4 == 0 by construction

    int mlim = Nl1 - 1;
    int mc0 = min(m_base + lm,      mlim);   // clamped: garbage rows dropped at store
    int mc1 = min(m_base + lm + 16, mlim);
    int mc2 = min(m_base + lm + 32, mlim);
    int mc3 = min(m_base + lm + 48, mlim);

    v8f acc0 = {}, acc1 = {}, acc2 = {}, acc3 = {};
#pragma unroll 2
    for (int k = k0; k < k1; k += 4) {
        int ka = k + half * 2;
        const float* p0 = P + (size_t)ka * Nl1;
        const float* p1 = p0 + Nl1;
        v2f b;
        b.x = h[(size_t)ka * C + n];
        b.y = h[(size_t)(ka + 1) * C + n];
        v2f a0, a1, a2, a3;
        a0.x = p0[mc0];  a0.y = p1[mc0];
        a1.x = p0[mc1];  a1.y = p1[mc1];
        a2.x = p0[mc2];  a2.y = p1[mc2];
        a3.x = p0[mc3];  a3.y = p1[mc3];
        __builtin_prefetch(p0 + (size_t)8 * Nl1 + mc0, 0, 1);  // global_prefetch_b8
        acc0 = wmma_f32x4(a0, b, acc0);
        acc1 = wmma_f32x4(a1, b, acc1);
        acc2 = wmma_f32x4(a2, b, acc2);
        acc3 = wmma_f32x4(a3, b, acc3);
    }
#pragma unroll
    for (int i = 0; i < 8; ++i) {
        int r  = i + half * 8;
        int m0 = m_base + r;
        if (m0      < Nl1) atomicAdd(&xn[(size_t)(m0)      * C + n], acc0[i]);
        if (m0 + 16 < Nl1) atomicAdd(&xn[(size_t)(m0 + 16) * C + n], acc1[i]);
        if (m0 + 32 < Nl1) atomicAdd(&xn[(size_t)(m0 + 32) * C + n], acc2[i]);
        if (m0 + 48 < Nl1) atomicAdd(&xn[(size_t)(m0 + 48) * C + n], acc3[i]);
    }
}

// ---------------------------------------------------------------------------
// out[c] = max over rows of x[N, C]
__global__ void max_over_nodes(const float* __restrict__ x, float* __restrict__ out,
                               int N, int C) {
    int c = blockIdx.x * blockDim.x + threadIdx.x;
    if (c >= C) return;
    float m = x[c];
    for (int i = 1; i < N; ++i) m = fmaxf(m, x[(size_t)i * C + c]);
    out[c] = m;
}

// ---------------------------------------------------------------------------
static inline int idiv_up(int a, int b) { return (a + b - 1) / b; }

extern "C" void kernel_launch(void* const* d_in, const int* in_sizes, int n_in,
                              void* d_out, int out_size, void* d_ws, size_t ws_size,
                              hipStream_t stream) {
    static const int NSa[7] = {20000, 5000, 1280, 320, 80, 32, 16};
    static const int ESa[6] = {320000, 80000, 20480, 5120, 1280, 512};
    static const int NFa[7] = {2, 16, 32, 64, 128, 128, 128};

    const float* x0 = (const float*)d_in[0];
    const int*   ei[6];
    const float* ea[6];
    const float* Pm[6];
    const float* Wm[6];
    const float* Rm[6];
    const float* bm[6];
    for (int l = 0; l < 6; ++l) {
        ei[l] = (const int*)  d_in[1 + 2 * l];
        ea[l] = (const float*)d_in[2 + 2 * l];
        Pm[l] = (const float*)d_in[13 + l];
        Wm[l] = (const float*)d_in[19 + 3 * l];
        Rm[l] = (const float*)d_in[20 + 3 * l];
        bm[l] = (const float*)d_in[21 + 3 * l];
    }

    // workspace partition (floats): A (max 10M), deg (20000), h (320000), x ping-pong
    float* wsf  = (float*)d_ws;
    float* Abuf = wsf;
    float* deg  = Abuf + 10000000;
    float* hbuf = deg  + 20000;
    float* xA   = hbuf + 320000;
    float* xB   = xA   + 80000;
    float* xbufs[2] = {xA, xB};

    const float* xcur = x0;
    for (int l = 0; l < 6; ++l) {
        int N = NSa[l], E = ESa[l], Cin = NFa[l], Cout = NFa[l + 1], N1 = NSa[l + 1];

        size_t Asz = (size_t)N * K3 * Cin;
        hipMemsetAsync(Abuf, 0, Asz * sizeof(float), stream);
        hipMemsetAsync(deg,  0, (size_t)N * sizeof(float), stream);

        degree_kernel<<<idiv_up(E, 256), 256, 0, stream>>>(ei[l], deg, E);
        scatter_edges<<<idiv_up(E * Cin, 256), 256, 0, stream>>>(xcur, ei[l], ea[l], Abuf, E, Cin);

        dim3 g1((unsigned)idiv_up(N, 16), (unsigned)(Cout / 16));
        spline_gemm<<<g1, 32, 0, stream>>>(Abuf, Wm[l], xcur, Rm[l], bm[l], deg, hbuf,
                                           N, Cin, Cout);

        float* xn = xbufs[l & 1];
        hipMemsetAsync(xn, 0, (size_t)N1 * Cout * sizeof(float), stream);
        int gx = idiv_up(N1, 64), gy = Cout / 16;
        int splitk = 4096 / (gx * gy);
        if (splitk < 1) splitk = 1;
        if (splitk > 64) splitk = 64;
        int kchunk = idiv_up(N, splitk * 4) * 4;       // multiple of 4
        splitk = idiv_up(N, kchunk);
        dim3 g2((unsigned)gx, (unsigned)gy, (unsigned)splitk);
        pool_gemm<<<g2, 32, 0, stream>>>(Pm[l], hbuf, xn, N, N1, Cout, kchunk);

        xcur = xn;
    }

    max_over_nodes<<<1, 128, 0, stream>>>(xcur, (float*)d_out, NSa[6], NFa[6]);
}